// weight_quantizer_fn_17927193493928
// MI455X (gfx1250) — compile-verified
//
#include <hip/hip_runtime.h>
#include <stdint.h>

// -------- CDNA5 async data-movement primitives (gfx1250) --------
// Per-lane async copy of 16B global -> LDS, tracked by ASYNCcnt.
// The LDS byte address is the low 32 bits of the flat address of a __shared__
// object (ISA 10.2: flat LDS aperture keeps the LDS offset in addr[31:0]).
__device__ __forceinline__ void async_load_b128_to_lds(void* ldsDst, const void* gSrc) {
    unsigned lds_addr = (unsigned)(uintptr_t)ldsDst;
    asm volatile("global_load_async_to_lds_b128 %0, %1, off"
                 :
                 : "v"(lds_addr), "v"(gSrc)
                 : "memory");
}
__device__ __forceinline__ void wait_async_le1() {
    asm volatile("s_wait_asynccnt 0x1" ::: "memory");
}
__device__ __forceinline__ void wait_async_le0() {
    asm volatile("s_wait_asynccnt 0x0" ::: "memory");
}

// -------- quantizer math (matches JAX reference numerics) --------
__device__ __forceinline__ float lsq_quant1(float w, float a) {
    // IEEE-correct division (default hipcc codegen), clamp to [-127,127],
    // round half-to-even (v_rndne_f32), rescale.
    float q = fminf(fmaxf(w / a, -127.0f), 127.0f);
    return rintf(q) * a;
}

// -------- main streaming kernel: async-LDS double buffered --------
// Requires: gridDim.x==GRID, blockDim.x==256, n4 == iters * stride4 exactly
// (fully uniform trip count -> EXEC all-ones throughout).
__global__ void __launch_bounds__(256)
lsq_quant_async(const float* __restrict__ w, const float* __restrict__ alpha,
                float* __restrict__ out, int iters, long long stride4) {
    __shared__ float4 smem[2][256];  // 8 KB of the 320 KB WGP LDS

    const float a = fmaxf(alpha[0], 1e-4f);
    const float4* __restrict__ w4 = (const float4*)w;
    float4* __restrict__ o4 = (float4*)out;

    const int tid = threadIdx.x;
    long long i = (long long)blockIdx.x * 256 + tid;

    // prologue: stage 0 in flight
    async_load_b128_to_lds(&smem[0][tid], (const void*)(w4 + i));

    int buf = 0;
    for (int it = 0; it < iters; ++it) {
        const long long nxt = i + stride4;
        if (it + 1 < iters) {
            // issue stage it+1, then wait until only it+1 is outstanding
            async_load_b128_to_lds(&smem[buf ^ 1][tid], (const void*)(w4 + nxt));
            wait_async_le1();
        } else {
            wait_async_le0();
        }

        const float4 v = smem[buf][tid];  // own slot only: no barrier needed
        float4 r;
        r.x = lsq_quant1(v.x, a);
        r.y = lsq_quant1(v.y, a);
        r.z = lsq_quant1(v.z, a);
        r.w = lsq_quant1(v.w, a);
        o4[i] = r;

        i = nxt;
        buf ^= 1;
    }
}

// -------- generic fallback (any n), grid-stride scalar --------
__global__ void __launch_bounds__(256)
lsq_quant_plain(const float* __restrict__ w, const float* __restrict__ alpha,
                float* __restrict__ out, long long n) {
    const float a = fmaxf(alpha[0], 1e-4f);
    const long long stride = (long long)gridDim.x * blockDim.x;
    for (long long i = (long long)blockIdx.x * blockDim.x + threadIdx.x; i < n; i += stride)
        out[i] = lsq_quant1(w[i], a);
}

// -------- bit-flip scatter fixup (runs after the main kernel) --------
// out[idx] = float(int32(trunc(clip(w/a))) XOR 0x80) * a
// Duplicated indices all compute the identical value -> benign write race,
// matching jnp .at[].set semantics on duplicates.
__global__ void __launch_bounds__(256)
lsq_flip(const float* __restrict__ w, const float* __restrict__ alpha,
         const int* __restrict__ idx, float* __restrict__ out, int m) {
    const int j = blockIdx.x * blockDim.x + threadIdx.x;
    if (j >= m) return;
    const float a = fmaxf(alpha[0], 1e-4f);
    const int i = idx[j];
    const float q = fminf(fmaxf(w[i] / a, -127.0f), 127.0f);
    const int t = (int)q;          // v_cvt_i32_f32: trunc toward zero == astype(int32)
    out[i] = (float)(t ^ 128) * a; // XOR MSB, then rescale (round of an int is itself)
}

extern "C" void kernel_launch(void* const* d_in, const int* in_sizes, int n_in,
                              void* d_out, int out_size, void* d_ws, size_t ws_size,
                              hipStream_t stream) {
    const float* w     = (const float*)d_in[0];
    const float* alpha = (const float*)d_in[1];
    const int*   flip  = (const int*)d_in[2];
    float*       out   = (float*)d_out;

    const long long n = (long long)in_sizes[0];   // 4096*8192 = 2^25
    const int nflip   = in_sizes[2];              // 3356

    constexpr int BLOCK = 256;                    // 8 wave32 per block
    constexpr int GRID  = 4096;
    const long long stride4 = (long long)GRID * BLOCK;  // 2^20 float4-threads
    const long long n4 = n >> 2;

    if (n > 0 && (n & 3LL) == 0 && (n4 % stride4) == 0) {
        const int iters = (int)(n4 / stride4);    // 8 for this shape
        lsq_quant_async<<<GRID, BLOCK, 0, stream>>>(w, alpha, out, iters, stride4);
    } else if (n > 0) {
        long long blocks = (n + BLOCK - 1) / BLOCK;
        if (blocks > 65536) blocks = 65536;
        lsq_quant_plain<<<(int)blocks, BLOCK, 0, stream>>>(w, alpha, out, n);
    }

    if (nflip > 0) {
        lsq_flip<<<(nflip + BLOCK - 1) / BLOCK, BLOCK, 0, stream>>>(w, alpha, flip, out, nflip);
    }
}